// PDPredictionGNN_8624294331203
// MI455X (gfx1250) — compile-verified
//
#include <hip/hip_runtime.h>

typedef __attribute__((ext_vector_type(2))) float v2f;
typedef __attribute__((ext_vector_type(8))) float v8f;
typedef __attribute__((ext_vector_type(4))) unsigned int u32x4;
typedef __attribute__((ext_vector_type(8))) int i32x8;
typedef __attribute__((ext_vector_type(4))) int i32x4;

#define HID 64
#define MAXK 128

static __device__ __forceinline__ v8f wmma_f32(v2f a, v2f b, v8f c) {
  return __builtin_amdgcn_wmma_f32_16x16x4_f32(false, a, false, b, (short)0, c, false, false);
}

// ---------------- degree / normalization ----------------

__global__ __launch_bounds__(256) void k_init_deg(float* __restrict__ deg, int n) {
  int i = blockIdx.x * blockDim.x + threadIdx.x;
  if (i < n) deg[i] = 1.0f;  // self-loop contribution
}

__global__ __launch_bounds__(256) void k_deg_accum(const long long* __restrict__ dst,
                                                   float* __restrict__ deg, int E) {
  int e = blockIdx.x * blockDim.x + threadIdx.x;
  if (e < E) atomicAdd(&deg[(int)dst[e]], 1.0f);
}

__global__ __launch_bounds__(256) void k_deg_to_dinv(float* __restrict__ deg, int n) {
  int i = blockIdx.x * blockDim.x + threadIdx.x;
  if (i < n) {
    float d = deg[i];
    deg[i] = d > 0.0f ? rsqrtf(d) : 0.0f;
  }
}

__global__ __launch_bounds__(256) void k_zero(float4* __restrict__ p, int n4) {
  int i = blockIdx.x * blockDim.x + threadIdx.x;
  if (i < n4) p[i] = make_float4(0.f, 0.f, 0.f, 0.f);
}

// ---------------- WMMA GEMM: H[M x 64] = X[M x K] @ W[K x 64] ----------------
// One wave computes a 32x64 output tile: eight 16x16 f32 accumulators,
// stepping K by 4 with V_WMMA_F32_16X16X4_F32 (each B fragment feeds 2 WMMAs).
// W is staged in LDS pair-interleaved: wlds[(k>>1)*128 + n*2 + (k&1)] = W[k][n],
// so the per-lane B fragment {W[k+2hi][n], W[k+2hi+1][n]} is one aligned ds_load_b64.
// Layouts (ISA 7.12.2, 32-bit):
//   A 16x4 : lane(lo,hi): v[j] = A[M=lo][K=2*hi+j]
//   B 4x16 : lane(lo,hi): v[j] = B[K=2*hi+j][N=lo]
//   C 16x16: VGPR r holds M = r + 8*hi, N = lo
__global__ __launch_bounds__(256) void k_gemm_wmma(const float* __restrict__ X,
                                                   const float* __restrict__ W,
                                                   float* __restrict__ H,
                                                   int M, int K) {
  __shared__ float wlds[(MAXK / 2) * 128];  // 32 KB max (K=128), pair-interleaved
  for (int i = threadIdx.x; i < K * HID; i += blockDim.x) {
    int k = i >> 6, n = i & 63;
    wlds[(k >> 1) * 128 + n * 2 + (k & 1)] = W[i];
  }
  __syncthreads();

  const int lane = threadIdx.x & 31;
  const int lo = lane & 15;
  const int hi = lane >> 4;

  const int task = blockIdx.x * (blockDim.x >> 5) + (threadIdx.x >> 5);
  const int numTasks = (M + 31) >> 5;  // 32 rows per wave
  if (task >= numTasks) return;
  const int m0 = task << 5;

  // clamp so a partial last tile still issues full-EXEC WMMA (junk masked at store)
  const int r0 = (m0 + lo < M) ? (m0 + lo) : (M - 1);
  const int r1 = (m0 + 16 + lo < M) ? (m0 + 16 + lo) : (M - 1);
  const float* __restrict__ x0 = X + (size_t)r0 * K;
  const float* __restrict__ x1 = X + (size_t)r1 * K;

  v8f c00 = {}, c01 = {}, c02 = {}, c03 = {};
  v8f c10 = {}, c11 = {}, c12 = {}, c13 = {};

  for (int k = 0; k < K; k += 4) {
    v2f a0 = *(const v2f*)&x0[k + 2 * hi];
    v2f a1 = *(const v2f*)&x1[k + 2 * hi];

    const v2f* __restrict__ wp = (const v2f*)&wlds[((k >> 1) + hi) * 128] + lo;
    v2f b0 = wp[0];    // N tile 0 (cols 0..15)
    v2f b1 = wp[16];   // N tile 1
    v2f b2 = wp[32];   // N tile 2
    v2f b3 = wp[48];   // N tile 3

    c00 = wmma_f32(a0, b0, c00);
    c01 = wmma_f32(a0, b1, c01);
    c02 = wmma_f32(a0, b2, c02);
    c03 = wmma_f32(a0, b3, c03);
    c10 = wmma_f32(a1, b0, c10);
    c11 = wmma_f32(a1, b1, c11);
    c12 = wmma_f32(a1, b2, c12);
    c13 = wmma_f32(a1, b3, c13);
  }

#pragma unroll
  for (int r = 0; r < 8; ++r) {
    int mrow = m0 + r + 8 * hi;
    if (mrow < M) {
      float* __restrict__ hp = H + (size_t)mrow * HID + lo;
      hp[0] = c00[r]; hp[16] = c01[r]; hp[32] = c02[r]; hp[48] = c03[r];
    }
    int mrow2 = m0 + 16 + r + 8 * hi;
    if (mrow2 < M) {
      float* __restrict__ hp = H + (size_t)mrow2 * HID + lo;
      hp[0] = c10[r]; hp[16] = c11[r]; hp[32] = c12[r]; hp[48] = c13[r];
    }
  }
}

// ---------------- edge gather + scatter-add (one wave per edge) ----------------
__global__ __launch_bounds__(256) void k_edge_agg(const float* __restrict__ h,
                                                  const long long* __restrict__ src,
                                                  const long long* __restrict__ dst,
                                                  const float* __restrict__ dinv,
                                                  float* __restrict__ agg, int E) {
  int task = blockIdx.x * (blockDim.x >> 5) + (threadIdx.x >> 5);
#if __has_builtin(__builtin_amdgcn_readfirstlane)
  int e = __builtin_amdgcn_readfirstlane(task);  // force scalar index path
#else
  int e = task;
#endif
  if (e >= E) return;
  int lane = threadIdx.x & 31;
  int s = (int)src[e];
  int d = (int)dst[e];
  float nrm = dinv[s] * dinv[d];
  const float2* __restrict__ hp = (const float2*)(h + (size_t)s * HID) + lane;
  float2 v = *hp;  // 256 B per wave, hot in L2 (25.6 MB table)
  float* __restrict__ ap = agg + (size_t)d * HID + lane * 2;
  atomicAdd(ap, v.x * nrm);      // global_atomic_add_f32, resolved in L2
  atomicAdd(ap + 1, v.y * nrm);
}

// ---------------- self-loop + bias + ReLU (in-place on agg) ----------------
__global__ __launch_bounds__(256) void k_finalize_relu(const float* __restrict__ h,
                                                       const float* __restrict__ dinv,
                                                       const float* __restrict__ b,
                                                       float* __restrict__ agg, int total) {
  int idx = blockIdx.x * blockDim.x + threadIdx.x;
  if (idx >= total) return;
  int i = idx >> 6;    // node
  int j = idx & 63;    // feature
  float di = dinv[i];
  float v = agg[idx] + h[idx] * di * di + b[j];
  agg[idx] = v > 0.0f ? v : 0.0f;
}

// ---------------- fused MLP head: relu(h@Wp1+bp1)@Wp2+bp2 ----------------
// Wp1 (8 KB contiguous) is staged into LDS by the Tensor Data Mover: wave 0
// issues one tensor_load_to_lds descriptor, waits TENSORcnt, then the block
// barrier publishes it. (6-arg clang-23 builtin form: extra i32x8 group.)
__global__ __launch_bounds__(256) void k_mlp(const float* __restrict__ h,
                                             const float* __restrict__ Wp1,
                                             const float* __restrict__ bp1,
                                             const float* __restrict__ Wp2,
                                             const float* __restrict__ bp2,
                                             float* __restrict__ out, int n) {
  __shared__ float w1s[HID * 32];
  __shared__ float b1s[32];
  __shared__ float w2s[32];
  __shared__ float b2s;

#if __has_builtin(__builtin_amdgcn_tensor_load_to_lds)
  if ((threadIdx.x >> 5) == 0) {  // one wave issues the DMA (TDM ignores EXEC)
    unsigned ldsOff = (unsigned)(size_t)(void*)w1s;          // LDS byte offset
    unsigned long long ga = (unsigned long long)(size_t)Wp1;  // global byte addr
    const unsigned elems = HID * 32;  // 2048 f32 elements
    u32x4 g0;
    g0[0] = 1u;                                               // count=1, user-mode
    g0[1] = ldsOff;                                           // lds_addr
    g0[2] = (unsigned)ga;                                     // global_addr[31:0]
    g0[3] = (unsigned)((ga >> 32) & 0x01FFFFFFull) | (2u << 30);  // addr[56:32] | type=2
    i32x8 g1;
    g1[0] = (int)(2u << 16);                 // data_size = 4 B
    g1[1] = (int)((elems & 0xFFFFu) << 16);  // tensor_dim0[15:0]
    g1[2] = (int)(elems >> 16) | (1 << 16);  // tensor_dim0[31:16] | tensor_dim1 = 1
    g1[3] = (int)(elems << 16);              // tile_dim0 = 2048 (1D tile)
    g1[4] = 0;                               // tile_dim1 = 0 (unused), tile_dim2 = 0
    g1[5] = (int)elems;                      // tensor_dim0_stride
    g1[6] = 0;
    g1[7] = 0;
    i32x4 g2 = {0, 0, 0, 0};
    i32x4 g3 = {0, 0, 0, 0};
    i32x8 g4 = {0, 0, 0, 0, 0, 0, 0, 0};
    __builtin_amdgcn_tensor_load_to_lds(g0, g1, g2, g3, g4, 0);
    __builtin_amdgcn_s_wait_tensorcnt(0);
  }
#else
  for (int i = threadIdx.x; i < HID * 32; i += blockDim.x) w1s[i] = Wp1[i];
#endif
  if (threadIdx.x < 32) {
    b1s[threadIdx.x] = bp1[threadIdx.x];
    w2s[threadIdx.x] = Wp2[threadIdx.x];
  }
  if (threadIdx.x == 0) b2s = bp2[0];
  __syncthreads();

  int i = blockIdx.x * blockDim.x + threadIdx.x;
  if (i >= n) return;

  float acc[32];
#pragma unroll
  for (int j = 0; j < 32; ++j) acc[j] = b1s[j];

  const float* __restrict__ row = h + (size_t)i * HID;
  for (int k = 0; k < HID; ++k) {
    float xk = row[k];
#pragma unroll
    for (int j = 0; j < 32; ++j) acc[j] = fmaf(xk, w1s[k * 32 + j], acc[j]);
  }

  float o = b2s;
#pragma unroll
  for (int j = 0; j < 32; ++j) {
    float r = acc[j] > 0.0f ? acc[j] : 0.0f;
    o = fmaf(r, w2s[j], o);
  }
  out[i] = o;
}

// ---------------- host orchestration ----------------
extern "C" void kernel_launch(void* const* d_in, const int* in_sizes, int n_in,
                              void* d_out, int out_size, void* d_ws, size_t ws_size,
                              hipStream_t stream) {
  (void)n_in; (void)ws_size; (void)out_size;
  const float*     x    = (const float*)d_in[0];
  const long long* ei   = (const long long*)d_in[1];   // edge_index is int64 in reference
  const float*     W1   = (const float*)d_in[3];
  const float*     b1   = (const float*)d_in[4];
  const float*     W2   = (const float*)d_in[5];
  const float*     b2   = (const float*)d_in[6];
  const float*     W3   = (const float*)d_in[7];
  const float*     b3   = (const float*)d_in[8];
  const float*     Wp1  = (const float*)d_in[9];
  const float*     bp1  = (const float*)d_in[10];
  const float*     Wp2  = (const float*)d_in[11];
  const float*     bp2  = (const float*)d_in[12];

  const int N = in_sizes[0] / 128;   // 100000
  const int E = in_sizes[1] / 2;     // 1600000
  const long long* srcI = ei;
  const long long* dstI = ei + E;

  float* ws   = (float*)d_ws;
  float* dinv = ws;                                   // N floats
  size_t nPad = ((size_t)N + 255) & ~(size_t)255;
  float* bufH = ws + nPad;                            // N*64 (GEMM output)
  float* bufA = bufH + (size_t)N * HID;               // N*64 (aggregation / layer output)

  const int T = 256;
  dim3 blk(T);

  // degree -> dinv
  k_init_deg<<<dim3((N + T - 1) / T), blk, 0, stream>>>(dinv, N);
  k_deg_accum<<<dim3((E + T - 1) / T), blk, 0, stream>>>(dstI, dinv, E);
  k_deg_to_dinv<<<dim3((N + T - 1) / T), blk, 0, stream>>>(dinv, N);

  const int tasks     = (N + 31) / 32;               // 32-row tiles per wave
  const int gemmBlks  = (tasks + (T / 32) - 1) / (T / 32);
  const int total     = N * HID;
  const int zeroBlks  = (total / 4 + T - 1) / T;
  const int edgeBlks  = ((size_t)E * 32 + T - 1) / T;  // one wave per edge
  const int finBlks   = (total + T - 1) / T;

  const float* layerIn = x;
  const float* Ws[3] = {W1, W2, W3};
  const float* bs[3] = {b1, b2, b3};
  int Ks[3] = {128, HID, HID};

  for (int l = 0; l < 3; ++l) {
    k_gemm_wmma<<<dim3(gemmBlks), blk, 0, stream>>>(layerIn, Ws[l], bufH, N, Ks[l]);
    k_zero<<<dim3(zeroBlks), blk, 0, stream>>>((float4*)bufA, total / 4);
    k_edge_agg<<<dim3(edgeBlks), blk, 0, stream>>>(bufH, srcI, dstI, dinv, bufA, E);
    k_finalize_relu<<<dim3(finBlks), blk, 0, stream>>>(bufH, dinv, bs[l], bufA, total);
    layerIn = bufA;
  }

  k_mlp<<<dim3((N + T - 1) / T), blk, 0, stream>>>(bufA, Wp1, bp1, Wp2, bp2,
                                                   (float*)d_out, N);
}